// Conv2dMap_40518721471188
// MI455X (gfx1250) — compile-verified
//
#include <hip/hip_runtime.h>

typedef __attribute__((ext_vector_type(2))) float v2f;
typedef __attribute__((ext_vector_type(8))) float v8f;

#define IN_C   8
#define OUT_C  16
#define KTAPS  9                 // 3x3
#define KTOT   (IN_C * KTAPS)    // 72
#define NCHUNK (KTOT / 4)        // 18 WMMA K-steps of 4
#define IH     512
#define IW     512
#define OH     510
#define OW     510
#define LDS_W  520               // 514 cols needed, padded
#define NTILE  8                 // 16-px tiles per wave (wave covers 128 cols)

// LDS byte offset of B element k; slab layout is [c*4 + dy][col] so the
// per-wave output-row select is a pure additive base (rowsel * LDS_W * 4).
__host__ __device__ constexpr int BOFF(int k) {
    return (((k / KTAPS) * 4 + (k % KTAPS) / 3) * LDS_W + (k % KTAPS) % 3) * 4;
}

__global__ __launch_bounds__(256)
void Conv2dMap_40518721471188_kernel(const float* __restrict__ x,
                                     const float* __restrict__ weight,
                                     const float* __restrict__ bias,
                                     float* __restrict__ out) {
    __shared__ float lds[IN_C * 4 * LDS_W];   // 8ch x 4 rows x 514 cols (~66.5 KB)
    __shared__ float wA[KTOT * OUT_C];        // dense scattered weight [k][m]

    const int tid  = threadIdx.x;
    const int lane = tid & 31;
    const int wave = tid >> 5;                // 0..7
    const bool hi  = lane >= 16;              // lane half -> K+2/K+3 rows
    const int npix = lane & 15;
    const int m    = lane & 15;

    const int blk = blockIdx.x;
    const int img = blk / (OH / 2);
    const int oy0 = (blk - img * (OH / 2)) * 2;     // block covers rows oy0, oy0+1

    // ---- Stage input slab with async direct-to-LDS b128 copies:
    //      32 (c*4+dy) rows x 128 float4 chunks, 16 per thread ----
    const float* __restrict__ xin = x + (size_t)img * IN_C * IH * IW;
    for (int i = tid; i < 32 * 128; i += 256) {
        int j  = (i & 127) << 2;                     // col (multiple of 4)
        int rc = i >> 7;                             // c*4 + dy
        const float* src =
            xin + ((size_t)(rc >> 2) * IH + oy0 + (rc & 3)) * IW + j;
        unsigned dst = (unsigned)(size_t)&lds[rc * LDS_W + j];  // LDS byte offset
        asm volatile("global_load_async_to_lds_b128 %0, %1, off"
                     :: "v"(dst), "v"(src) : "memory");
    }
    // Tail cols 512,513: clamp to col 511 (feed only predicated-off outputs)
    if (tid < 64) {
        int rc = tid >> 1;
        lds[rc * LDS_W + 512 + (tid & 1)] =
            xin[((size_t)(rc >> 2) * IH + oy0 + (rc & 3)) * IW + (IW - 1)];
    }

    // ---- Build dense scattered weight table wA[k][16] (once per block) ----
    for (int i = tid; i < KTOT * OUT_C; i += 256) {  // 4.5 iters/thread
        int mm = i & 15;
        int k  = i >> 4;
        int c  = k / KTAPS;
        int t  = k - c * KTAPS;
        int d  = (c - mm) & 7;                       // in set iff d < 4
        float v = 0.0f;
        if (d < 4) {
            int r0 = 0;                              // rank (torch nonzero order)
#pragma unroll
            for (int r = 0; r < 4; ++r) r0 += (((mm + r) & 7) < c) ? 1 : 0;
            v = weight[(mm * 4 + r0) * KTAPS + t];
        }
        wA[(k << 4) + mm] = v;
    }

    // All waves drain their own async copies, then barrier -> slab visible.
    asm volatile("s_wait_asynccnt 0" ::: "memory");
    __syncthreads();

    // ---- Main loop: 18 K-chunks x 8 N-tiles = 144 WMMAs per wave ----
    const int rowsel = wave >> 2;                    // output row within block
    const int wg     = wave & 3;                     // 128-col group within row
    const float* wAp = wA + m + (hi ? 32 : 0);       // A: (k,m) at k*16+m
    // Per-wave B base: tile displacements are compile-time (t*64 bytes) so
    // every tile load folds into a ds_load immediate offset.
    const char* base0 =
        (const char*)lds + (rowsel * LDS_W + wg * 128 + npix) * 4;

    v8f acc[NTILE] = {};
#pragma unroll
    for (int kb = 0; kb < NCHUNK; ++kb) {
        v2f a;
        a[0] = wAp[(4 * kb + 0) * 16];               // ds_load, imm offsets
        a[1] = wAp[(4 * kb + 1) * 16];
        // B offsets: one cndmask of inline constants each, shared by 8 tiles
        int off0 = hi ? BOFF(4 * kb + 2) : BOFF(4 * kb + 0);
        int off1 = hi ? BOFF(4 * kb + 3) : BOFF(4 * kb + 1);
        const char* p0 = base0 + off0;               // one v_add each
        const char* p1 = base0 + off1;
#pragma unroll
        for (int t = 0; t < NTILE; ++t) {
            v2f b;
            b[0] = *(const float*)(p0 + t * 64);     // imm-offset ds loads
            b[1] = *(const float*)(p1 + t * 64);
            acc[t] = __builtin_amdgcn_wmma_f32_16x16x4_f32(
                false, a, false, b, (short)0, acc[t], false, false);
        }
    }

    // ---- Epilogue: bias + predicated coalesced stores ----
    const int ocbase = hi ? 8 : 0;
    const int oy     = oy0 + rowsel;
    float* __restrict__ outp =
        out + (size_t)img * OUT_C * (OH * OW) + (size_t)oy * OW;
#pragma unroll
    for (int t = 0; t < NTILE; ++t) {
        int ox = wg * 128 + t * 16 + npix;
        if (ox < OW) {
#pragma unroll
            for (int r = 0; r < 8; ++r) {
                int ch = ocbase + r;
                outp[(size_t)ch * (OH * OW) + ox] = acc[t][r] + bias[ch];
            }
        }
    }
}

extern "C" void kernel_launch(void* const* d_in, const int* in_sizes, int n_in,
                              void* d_out, int out_size, void* d_ws, size_t ws_size,
                              hipStream_t stream) {
    const float* x      = (const float*)d_in[0];
    const float* weight = (const float*)d_in[1];
    const float* bias   = (const float*)d_in[2];
    float* out          = (float*)d_out;

    dim3 grid(32 * (OH / 2));    // one block per (image, output-row pair)
    dim3 block(256);             // 8 wave32
    Conv2dMap_40518721471188_kernel<<<grid, block, 0, stream>>>(x, weight, bias, out);
}